// MSRSA_79482664780188
// MI455X (gfx1250) — compile-verified
//
#include <hip/hip_runtime.h>

typedef float v2f __attribute__((ext_vector_type(2)));
typedef float v8f __attribute__((ext_vector_type(8)));

#define SEQ   1024
#define DHEAD 64
#define NH    8
#define SROW  1028   // padded LDS row stride in floats (bank-conflict free)

__global__ __launch_bounds__(128) void msrsa_attn_kernel(
    const float* __restrict__ Q, const float* __restrict__ K,
    const float* __restrict__ V, const int* __restrict__ mask,
    const float* __restrict__ Dmat, const float* __restrict__ wD,
    float* __restrict__ out, float* __restrict__ attnw)
{
    __shared__ float S[16 * SROW];   // 16 q-rows x 1024 k (padded)
    __shared__ float red[128];
    __shared__ float rowstat[16];

    const int qt = blockIdx.x;               // 0..63  (q tile of 16)
    const int h  = blockIdx.y;               // 0..7
    const int b  = blockIdx.z;               // 0..3
    const int bh = b * NH + h;
    const int qbase = qt * 16;

    const int tid  = threadIdx.x;
    const int wave = tid >> 5;               // 0..3
    const int lane = tid & 31;
    const int m    = lane & 15;              // row (A) / col (B,C) index
    const int hi   = lane >> 4;              // 0 or 1
    const int koff = hi * 2;                 // contraction sub-offset per ISA layout

    const float w_d = wD[h];

    // ---------- preload Q A-fragments: 16 K=4 steps covering d=0..63 ----------
    const float* qrow = Q + ((size_t)bh * SEQ + (qbase + m)) * DHEAD;
    v2f qa[16];
    #pragma unroll
    for (int s = 0; s < 16; ++s)
        qa[s] = *(const v2f*)(qrow + 4 * s + koff);

    // ---------- Phase 1: S = Q K^T (masked + distance bias) into LDS ----------
    for (int t = 0; t < 16; ++t) {
        const int kt    = wave * 16 + t;     // 16x16 k tile index, 0..63
        const int kbase = kt * 16;
        const float* krow = K + ((size_t)bh * SEQ + (kbase + m)) * DHEAD;

        v8f c = {};
        #pragma unroll
        for (int s = 0; s < 16; ++s) {
            v2f kb = *(const v2f*)(krow + 4 * s + koff);   // B[d][n] = K[n][d]
            c = __builtin_amdgcn_wmma_f32_16x16x4_f32(
                    false, qa[s], false, kb, (short)0, c, false, false);
        }

        const int kcol = kbase + m;
        const int mok  = mask[b * SEQ + kcol];
        #pragma unroll
        for (int r = 0; r < 8; ++r) {
            const int qr = r + hi * 8;       // C layout: VGPR r -> row r / r+8
            float sv = 0.0f;
            if (mok > 0) {
                const int q = qbase + qr;
                float dv = Dmat[((size_t)b * SEQ + q) * SEQ + kcol];
                sv = c[r] * (1.0f + dv * w_d) * 0.125f;   // (S + S*D*wD)/sqrt(64)
            }
            S[qr * SROW + kcol] = sv;
        }
    }
    __syncthreads();

    // ---------- Phase 2: row softmax over k=1024 ----------
    const int row = tid >> 3;                // 0..15
    const int sub = tid & 7;                 // 8 threads per row

    float mx = -3.402823466e38f;
    for (int c = sub; c < SEQ; c += 8)
        mx = fmaxf(mx, S[row * SROW + c]);
    red[tid] = mx;
    __syncthreads();
    if (tid < 16) {
        float v = red[tid * 8];
        #pragma unroll
        for (int i = 1; i < 8; ++i) v = fmaxf(v, red[tid * 8 + i]);
        rowstat[tid] = v;
    }
    __syncthreads();

    const float rmax = rowstat[row];
    float sum = 0.0f;
    for (int c = sub; c < SEQ; c += 8) {
        float e = __expf(S[row * SROW + c] - rmax);
        S[row * SROW + c] = e;
        sum += e;
    }
    __syncthreads();
    red[tid] = sum;
    __syncthreads();
    if (tid < 16) {
        float v = 0.0f;
        #pragma unroll
        for (int i = 0; i < 8; ++i) v += red[tid * 8 + i];
        rowstat[tid] = 1.0f / v;
    }
    __syncthreads();

    // normalize in LDS + coalesced global write of attn_weight
    for (int r = 0; r < 16; ++r) {
        const float inv = rowstat[r];
        const size_t awbase = ((size_t)bh * SEQ + (qbase + r)) * SEQ;
        for (int c = tid; c < SEQ; c += 128) {
            float p = S[r * SROW + c] * inv;
            S[r * SROW + c] = p;
            attnw[awbase + c] = p;
        }
    }
    __syncthreads();

    // ---------- Phase 3: out = P V  (each wave owns one 16-wide d tile) ----------
    const int dt = wave;                     // d tile 0..3
    const float* vbase = V + (size_t)bh * SEQ * DHEAD + dt * 16 + m;

    v8f cacc = {};
    for (int kk = 0; kk < SEQ; kk += 4) {
        v2f a = *(const v2f*)&S[m * SROW + kk + koff];          // P fragment (LDS)
        v2f bf;
        bf.x = vbase[(size_t)(kk + koff) * DHEAD];              // B[k'][n] = V[k][d]
        bf.y = vbase[(size_t)(kk + koff + 1) * DHEAD];
        cacc = __builtin_amdgcn_wmma_f32_16x16x4_f32(
                   false, a, false, bf, (short)0, cacc, false, false);
    }

    #pragma unroll
    for (int r = 0; r < 8; ++r) {
        const int q = qbase + r + hi * 8;
        out[((size_t)bh * SEQ + q) * DHEAD + dt * 16 + m] = cacc[r];
    }
}

extern "C" void kernel_launch(void* const* d_in, const int* in_sizes, int n_in,
                              void* d_out, int out_size, void* d_ws, size_t ws_size,
                              hipStream_t stream) {
    const float* Q    = (const float*)d_in[0];
    const float* K    = (const float*)d_in[1];
    const float* V    = (const float*)d_in[2];
    const int*   mask = (const int*)d_in[3];
    const float* Dmat = (const float*)d_in[4];
    const float* wD   = (const float*)d_in[5];

    float* out = (float*)d_out;                          // (4,8,1024,64)
    float* aw  = out + (size_t)4 * NH * SEQ * DHEAD;     // (4,8,1024,1024)

    dim3 grid(SEQ / 16, NH, 4);
    msrsa_attn_kernel<<<grid, 128, 0, stream>>>(Q, K, V, mask, Dmat, wD, out, aw);
}